// ParallelE8Quantizer_43224550867170
// MI455X (gfx1250) — compile-verified
//
#include <hip/hip_runtime.h>

typedef __attribute__((ext_vector_type(2))) float v2f;
typedef __attribute__((ext_vector_type(8))) float v8f;

#define NUM_LEVELS 4
#define NROOTS     240
#define NTILES     15          // 240 / 16
#define NVEC       524288
#define RSTRIDE    18          // u64 stride per row: 36 dwords -> conflict-free

// One wave32 handles 16 vectors. A-matrix (16x4 f32) layout:
//   lanes 0-15:  M = lane,    VGPR0 = K0, VGPR1 = K1
//   lanes 16-31: M = lane-16, VGPR0 = K2, VGPR1 = K3
// => lane holds components {kb, kb+1} (chunk lo, K=0..3) and {kb+4, kb+5}
//    (chunk hi, K=4..7) of row (lane&15), where kb = 2*(lane>>4).
// B-matrix (4x16 f32) mirrors this with N = lane&15, rows = roots.
// C/D (16x16 f32): element (M,N) in lane (N + 16*(M>=8)), VGPR M%8.
//
// All 240 E8 roots have |root|^2 == 2, so argmin ||r - root||^2 == argmax r.root:
// the distance computation is a pure f32 GEMM on v_wmma_f32_16x16x4_f32.

__global__ void __launch_bounds__(256) e8q_kernel(
    const float* __restrict__ x,      // [NVEC, 8]
    const float* __restrict__ roots,  // [240, 8]
    float*       __restrict__ out_q,  // [NVEC, 8]
    int*         __restrict__ out_idx,// [4, NVEC]
    float*       __restrict__ ws_acc) // scalar accumulator
{
    __shared__ __align__(16) float sroots[NROOTS * 8];
    // Per-wave argmax transpose scratch: [wave][row 0..15][col 0..15] packed
    // (score,idx) pairs, row stride RSTRIDE u64 to spread LDS banks.
    __shared__ unsigned long long sred[8][16 * RSTRIDE];

#pragma unroll
    for (int i = 0; i < (NROOTS * 8 + 255) / 256; ++i) {
        int idx = i * 256 + threadIdx.x;
        if (idx < NROOTS * 8) sroots[idx] = roots[idx];
    }
    __syncthreads();

    const int lane     = threadIdx.x & 31;
    const int wib      = threadIdx.x >> 5;       // wave in block
    const int wave     = (blockIdx.x * blockDim.x + threadIdx.x) >> 5;
    const int base_row = wave << 4;              // 16 vectors per wave
    const int n        = lane & 15;              // column / row-in-tile id
    const int kb       = (lane >> 4) << 1;       // 0 or 2

    // Write base: this lane writes rows (lane>=16 ? 8.. : 0..)+j at column n.
    unsigned long long* const wp = &sred[wib][(lane >> 4) * 8 * RSTRIDE + n];
    // Read base: this lane reduces its own row r = lane & 15.
    const unsigned long long* const rp = &sred[wib][n * RSTRIDE];

    // B tiles resident in registers for all 4 levels: roots.T, 15 tiles of 8x16.
    v2f Blo[NTILES], Bhi[NTILES];
#pragma unroll
    for (int t = 0; t < NTILES; ++t) {
        const float* rps = &sroots[(16 * t + n) * 8 + kb];
        Blo[t] = *(const v2f*)(rps);
        Bhi[t] = *(const v2f*)(rps + 4);
    }

    // x stays resident; residual a = x - qsum recomputed like the reference.
    const float* xp = &x[(size_t)(base_row + n) * 8 + kb];
    const v2f xlo = *(const v2f*)(xp);
    const v2f xhi = *(const v2f*)(xp + 4);
    v2f alo = xlo, ahi = xhi;
    v2f qlo = {0.f, 0.f}, qhi = {0.f, 0.f};

    int sel[NUM_LEVELS];

#pragma unroll
    for (int level = 0; level < NUM_LEVELS; ++level) {
        float best[8];
        int   bidx[8];

        // Tile 0 initializes the running (max, argmax) -- no sentinel needed.
        {
            v8f c = {};
            c = __builtin_amdgcn_wmma_f32_16x16x4_f32(
                    false, alo, false, Blo[0], (short)0, c, false, false);
            c = __builtin_amdgcn_wmma_f32_16x16x4_f32(
                    false, ahi, false, Bhi[0], (short)0, c, false, false);
#pragma unroll
            for (int j = 0; j < 8; ++j) { best[j] = c[j]; bidx[j] = n; }
        }
#pragma unroll
        for (int t = 1; t < NTILES; ++t) {
            v8f c = {};
            c = __builtin_amdgcn_wmma_f32_16x16x4_f32(
                    false, alo, false, Blo[t], (short)0, c, false, false);
            c = __builtin_amdgcn_wmma_f32_16x16x4_f32(
                    false, ahi, false, Bhi[t], (short)0, c, false, false);
            const int colidx = 16 * t + n;
#pragma unroll
            for (int j = 0; j < 8; ++j) {
                const bool take = c[j] > best[j];   // strict > : first max wins
                best[j] = take ? c[j]    : best[j];
                bidx[j] = take ? colidx  : bidx[j];
            }
        }

        // Transpose partials through LDS: lane writes its 8 rows' (score,idx)
        // at column n; then reduces the 16 partials of its own row locally.
        // Scratch is per-wave and same-wave LDS ops are in-order: no barrier.
#pragma unroll
        for (int j = 0; j < 8; ++j) {
            const unsigned long long p =
                ((unsigned long long)(unsigned)bidx[j] << 32) |
                (unsigned long long)__float_as_uint(best[j]);
            wp[j * RSTRIDE] = p;
        }

        float vs[16];
        int   vi[16];
#pragma unroll
        for (int c = 0; c < 16; ++c) {
            const unsigned long long p = rp[c];
            vs[c] = __uint_as_float((unsigned)p);
            vi[c] = (int)(p >> 32);
        }
        // Lexicographic tree max on (score, -idx): ties -> lowest column index,
        // matching jnp.argmin's first-hit rule; order-independent.
#pragma unroll
        for (int w = 1; w < 16; w <<= 1) {
#pragma unroll
            for (int c = 0; c < 16; c += 2 * w) {
                const bool take = (vs[c + w] > vs[c]) ||
                                  (vs[c + w] == vs[c] && vi[c + w] < vi[c]);
                vs[c] = take ? vs[c + w] : vs[c];
                vi[c] = take ? vi[c + w] : vi[c];
            }
        }
        const int s = vi[0];          // winner for row (lane & 15)
        sel[level] = s;

        // Gather winning root from LDS; qsum += root; residual = x - qsum
        // (same update order as the reference).
        const float* sp = &sroots[s * 8 + kb];
        const v2f rlo = *(const v2f*)(sp);
        const v2f rhi = *(const v2f*)(sp + 4);
        qlo += rlo;  qhi += rhi;
        alo = xlo - qlo;  ahi = xhi - qhi;
    }

    // Quantized output (same component layout as A).
    float* op = &out_q[(size_t)(base_row + n) * 8 + kb];
    *(v2f*)(op)     = qlo;
    *(v2f*)(op + 4) = qhi;

    // Indices: lanes 0-15 own rows == lane; single predicated block.
    if (lane < 16) {
#pragma unroll
        for (int level = 0; level < NUM_LEVELS; ++level)
            out_idx[(size_t)level * NVEC + base_row + lane] = sel[level];
    }

    // qerr partial: final residual == x - quantized.
    float e = alo.x * alo.x + alo.y * alo.y + ahi.x * ahi.x + ahi.y * ahi.y;
#pragma unroll
    for (int st = 1; st < 32; st <<= 1) e += __shfl_xor(e, st, 32);
    if (lane == 0) atomicAdd(ws_acc, e);
}

__global__ void e8q_finalize(const float* __restrict__ ws_acc,
                             float* __restrict__ out_qerr)
{
    *out_qerr = *ws_acc * (1.0f / (float)((size_t)NVEC * 8));
}

extern "C" void kernel_launch(void* const* d_in, const int* in_sizes, int n_in,
                              void* d_out, int out_size, void* d_ws, size_t ws_size,
                              hipStream_t stream)
{
    const float* x     = (const float*)d_in[0];   // [NVEC, 8] f32
    const float* roots = (const float*)d_in[1];   // [240, 8]  f32

    float* out_q    = (float*)d_out;                                   // NVEC*8 f32
    int*   out_idx  = (int*)((float*)d_out + (size_t)NVEC * 8);        // 4*NVEC i32
    float* out_qerr = (float*)d_out + (size_t)NVEC * 8
                                    + (size_t)NUM_LEVELS * NVEC;       // 1 f32

    float* acc = (float*)d_ws;
    hipMemsetAsync(acc, 0, sizeof(float), stream);

    // 16 vectors per wave, 8 waves per block -> 128 vectors per block.
    dim3 block(256);
    dim3 grid(NVEC / 128);
    e8q_kernel<<<grid, block, 0, stream>>>(x, roots, out_q, out_idx, acc);
    e8q_finalize<<<1, 1, 0, stream>>>(acc, out_qerr);
}